// CoreDecoderStatefull_53721450939074
// MI455X (gfx1250) — compile-verified
//
#include <hip/hip_runtime.h>

// ---------------------------------------------------------------------------
// Persistent single-workgroup decoder for MI455X (gfx1250, wave32).
// Every mat-vec runs on the matrix pipe via V_WMMA_F32_16X16X4_F32 with the
// WEIGHT tile in the A fragment (ISA 32-bit A 16x4 layout: lane m holds
// W[nb+m][kb+0..1], lane m+16 holds W[nb+m][kb+2..3] -> contiguous b64 loads)
// and the x-vector as column 0 of the B fragment (broadcast LDS reads on
// lanes 0/16).  D column 0 = 16 outputs, resident on lanes 0 (M=0..7) and
// 16 (M=8..15) across the 8 accumulator VGPRs.
// Activations live in LDS; weights (3.6 MB fp32 total) stay resident in the
// 192 MB L2 across graph replays.  Next stage's GRU input weights are
// prefetched (global_prefetch_b8) while the current stage computes, hiding
// part of the serial dependency chain.
// ---------------------------------------------------------------------------

typedef float v2f __attribute__((ext_vector_type(2)));
typedef float v8f __attribute__((ext_vector_type(8)));

__device__ __forceinline__ float clamp1(float v) {
    return fminf(fmaxf(v, -1.0f), 1.0f);
}
__device__ __forceinline__ float sigm(float v) {
    return 1.0f / (1.0f + __expf(-v));
}

// Accumulate a 16-output tile:  out[nb..nb+15] += W(rows nb..nb+15) . x
// W row-major, row stride ldw, element stride along K = ks.  kLen % 4 == 0.
//   A[m][k] = W[nb+m][kb+k]   (per-lane consecutive pair when ks==1 -> b64)
//   B[k][0] = x[kb+k]         (only column 0 non-zero)
__device__ __forceinline__ v8f wmma_mv(v8f c, const float* __restrict__ W,
                                       int ldw, int ks,
                                       const float* __restrict__ x,
                                       int kLen, int nb, int lane) {
    const int nIdx = lane & 15;    // row of the weight tile this lane carries
    const int kh   = lane >> 4;    // which K-half (0: K=0,1  1: K=2,3)
    const float* wrow = W + (nb + nIdx) * ldw;
    for (int kb = 0; kb < kLen; kb += 4) {
        v2f a, b;
        // A fragment: two consecutive K elements of this lane's weight row.
        a.x = wrow[(kb + 2 * kh) * ks];
        a.y = wrow[(kb + 2 * kh + 1) * ks];
        // B fragment: x in column 0 only.
        //   vgpr0: lane0 = x[kb+0], lane16 = x[kb+1]
        //   vgpr1: lane0 = x[kb+2], lane16 = x[kb+3]
        float xb0 = x[kb + kh];
        float xb1 = x[kb + 2 + kh];
        b.x = (nIdx == 0) ? xb0 : 0.0f;
        b.y = (nIdx == 0) ? xb1 : 0.0f;
        c = __builtin_amdgcn_wmma_f32_16x16x4_f32(false, a, false, b,
                                                  (short)0, c, false, false);
    }
    return c;
}

// D column 0 lives on lane 0 (M=0..7) and lane 16 (M=8..15).
__device__ __forceinline__ void store_tile(v8f c, float* __restrict__ out,
                                           const float* __restrict__ bias,
                                           int nb, int lane) {
    if ((lane & 15) == 0) {
        const int base = nb + (lane >> 4) * 8;
#pragma unroll
        for (int r = 0; r < 8; ++r) {
            float v = c[r];
            if (bias) v += bias[base + r];
            out[base + r] = v;
        }
    }
}

// Block-wide gemv: nOut outputs (multiple of 16) spread over the 8 waves.
__device__ __forceinline__ void block_mv(const float* __restrict__ W, int ldw,
                                         const float* __restrict__ x, int kLen,
                                         int nOut, float* __restrict__ out,
                                         const float* __restrict__ bias) {
    const int lane = threadIdx.x & 31;
    const int wave = threadIdx.x >> 5;
    const int nw   = blockDim.x >> 5;
    for (int nb = wave * 16; nb < nOut; nb += nw * 16) {
        v8f c = {};
        c = wmma_mv(c, W, ldw, 1, x, kLen, nb, lane);
        store_tile(c, out, bias, nb, lane);
    }
}

struct Params {
    const float* z;
    const float* h[5];
    const float* c[5];
    const float* w_dense;
    const float* b_dense;
    const float* g_wi[5];
    const float* g_wh[5];
    const float* g_bi[5];
    const float* g_bh[5];
    const float* cv_w[5];
    const float* cv_b[5];
    const float* glu_w[5];
    const float* w_out;
    const float* b_out;
    float* out;
};

__global__ void __launch_bounds__(256, 1)
core_decoder_kernel(Params p) {
    __shared__ float xls[736];    // growing activation vector
    __shared__ float hls[96];     // current stage hidden state
    __shared__ float cls[704];    // current stage conv memory (cprev)
    __shared__ float gils[288];   // GRU input gates
    __shared__ float ghls[288];   // GRU hidden gates
    __shared__ float hnls[96];    // new hidden state
    __shared__ float tls[96];     // temp (dense pre-act / GLU gate)
    __shared__ float convls[32];  // conv pre-activation
    __shared__ float zls[80];
    __shared__ float outls[80];

    const int tid = threadIdx.x;

    // ---- input dense: x = clamp(tanh(W_dense @ z + b_dense)) --------------
    for (int i = tid; i < 80; i += 256) zls[i] = p.z[i];
    // warm L2 for the first GRU weights while z lands in LDS
    for (int i = tid * 32; i < 288 * 96; i += 256 * 32)
        __builtin_prefetch(p.g_wi[0] + i, 0, 3);
    __syncthreads();

    block_mv(p.w_dense, 80, zls, 80, 96, tls, p.b_dense);
    __syncthreads();
    for (int i = tid; i < 96; i += 256) xls[i] = clamp1(tanhf(tls[i]));
    __syncthreads();

    // ---- 5 stages: GRU -> GLU -> concat -> stateful conv -> concat --------
    int gd = 96;
    for (int s = 0; s < 5; ++s) {
        const int cd = gd + 96;

        for (int i = tid; i < 96; i += 256) hls[i] = p.h[s][i];
        for (int i = tid; i < cd; i += 256) cls[i] = p.c[s][i];

        // prefetch next stage's (largest) weight matrix into L2/WGP$
        if (s + 1 < 5) {
            const float* nw = p.g_wi[s + 1];
            const int nfl = 288 * (gd + 128);
            for (int i = tid * 32; i < nfl; i += 256 * 32)
                __builtin_prefetch(nw + i, 0, 3);
        }
        __syncthreads();

        // GRU gate gemvs (wave-parallel, no barrier needed between the two)
        block_mv(p.g_wi[s], gd, xls, gd, 288, gils, p.g_bi[s]);
        block_mv(p.g_wh[s], 96, hls, 96, 288, ghls, p.g_bh[s]);
        __syncthreads();

        // gate nonlinearity (PyTorch order r,z,n) + clamp (noise approx)
        for (int i = tid; i < 96; i += 256) {
            float r  = sigm(gils[i]        + ghls[i]);
            float u  = sigm(gils[96 + i]   + ghls[96 + i]);
            float cg = tanhf(gils[192 + i] + r * ghls[192 + i]);
            hnls[i] = clamp1((1.0f - u) * cg + u * hls[i]);
        }
        __syncthreads();

        // GLU: g = clamp(hn * sigmoid(glu_w @ hn))
        block_mv(p.glu_w[s], 96, hnls, 96, 96, tls, nullptr);
        __syncthreads();
        for (int i = tid; i < 96; i += 256)
            xls[gd + i] = clamp1(hnls[i] * sigm(tls[i]));
        __syncthreads();

        // stateful conv1d (k=2): tap0 over cprev, tap1 over current x_cat.
        // cv_w layout (32, cd, 2): row stride 2*cd, K-stride 2, tap = +0/+1.
        {
            const int lane = tid & 31;
            const int wave = tid >> 5;
            for (int nb = wave * 16; nb < 32; nb += 8 * 16) {
                v8f cacc = {};
                cacc = wmma_mv(cacc, p.cv_w[s] + 0, 2 * cd, 2, cls, cd, nb, lane);
                cacc = wmma_mv(cacc, p.cv_w[s] + 1, 2 * cd, 2, xls, cd, nb, lane);
                store_tile(cacc, convls, p.cv_b[s], nb, lane);
            }
        }
        __syncthreads();
        for (int i = tid; i < 32; i += 256)
            xls[cd + i] = clamp1(tanhf(convls[i]));
        __syncthreads();

        gd = cd + 32;   // 96 -> 224 -> 352 -> 480 -> 608 -> 736
    }

    // ---- output projection: y = W_out @ x + b_out  (80 outs, K=736) -------
    block_mv(p.w_out, 736, xls, 736, 80, outls, p.b_out);
    __syncthreads();
    for (int i = tid; i < 80; i += 256) p.out[i] = outls[i];
}

extern "C" void kernel_launch(void* const* d_in, const int* in_sizes, int n_in,
                              void* d_out, int out_size, void* d_ws, size_t ws_size,
                              hipStream_t stream) {
    (void)in_sizes; (void)n_in; (void)out_size; (void)d_ws; (void)ws_size;

    Params p;
    int k = 0;
    p.z = (const float*)d_in[k++];                       // 0
    for (int i = 0; i < 5; ++i) p.h[i] = (const float*)d_in[k++];   // 1..5
    for (int i = 0; i < 5; ++i) p.c[i] = (const float*)d_in[k++];   // 6..10
    p.w_dense = (const float*)d_in[k++];                 // 11
    p.b_dense = (const float*)d_in[k++];                 // 12
    for (int i = 0; i < 5; ++i) {                        // 13..32
        p.g_wi[i] = (const float*)d_in[k++];
        p.g_wh[i] = (const float*)d_in[k++];
        p.g_bi[i] = (const float*)d_in[k++];
        p.g_bh[i] = (const float*)d_in[k++];
    }
    for (int i = 0; i < 5; ++i) {                        // 33..42
        p.cv_w[i] = (const float*)d_in[k++];
        p.cv_b[i] = (const float*)d_in[k++];
    }
    for (int i = 0; i < 5; ++i) p.glu_w[i] = (const float*)d_in[k++]; // 43..47
    p.w_out = (const float*)d_in[k++];                   // 48
    p.b_out = (const float*)d_in[k++];                   // 49
    p.out   = (float*)d_out;

    core_decoder_kernel<<<1, 256, 0, stream>>>(p);
}